// GPRGNN_83794811945393
// MI455X (gfx1250) — compile-verified
//
#include <hip/hip_runtime.h>
#include <hip/hip_bf16.h>

typedef __attribute__((ext_vector_type(2))) float v2f;
typedef __attribute__((ext_vector_type(8))) float v8f;

#define N_NODES 100000
#define N_EDGES 1600000
#define IN_DIM 256
#define HID 256
#define OUT_DIM 64
#define K_STEPS 10
#define LDS_STRIDE 258   // 258 % 64 == 2 -> 16 rows land on 16 distinct banks, 8B aligned

// ---------------- softmax over 11 alpha values -> gamma ----------------
__global__ void gamma_kernel(const float* __restrict__ alpha, float* __restrict__ gamma) {
    if (threadIdx.x == 0) {
        float m = -1e30f;
        for (int i = 0; i <= K_STEPS; ++i) m = fmaxf(m, alpha[i]);
        float e[K_STEPS + 1], s = 0.f;
        for (int i = 0; i <= K_STEPS; ++i) { e[i] = __expf(alpha[i] - m); s += e[i]; }
        for (int i = 0; i <= K_STEPS; ++i) gamma[i] = e[i] / s;
    }
}

// ---------------- degree / norm ----------------
__global__ void zero_f32(float* __restrict__ p, long long n) {
    long long i = (long long)blockIdx.x * blockDim.x + threadIdx.x;
    if (i < n) p[i] = 0.f;
}

__global__ void degree_kernel(const long long* __restrict__ row, float* __restrict__ deg, int e_cnt) {
    int i = blockIdx.x * blockDim.x + threadIdx.x;
    if (i < e_cnt) atomicAdd(&deg[(int)row[i]], 1.0f);
}

__global__ void dinvsqrt_kernel(float* __restrict__ deg, int n) {
    int i = blockIdx.x * blockDim.x + threadIdx.x;
    if (i < n) {
        float d = deg[i];
        deg[i] = (d > 0.f) ? rsqrtf(fmaxf(d, 1.0f)) : 0.f;
    }
}

__global__ void norm_kernel(const long long* __restrict__ row, const long long* __restrict__ col,
                            const float* __restrict__ dis, float* __restrict__ norm, int e_cnt) {
    int i = blockIdx.x * blockDim.x + threadIdx.x;
    if (i < e_cnt) norm[i] = dis[(int)row[i]] * dis[(int)col[i]];
}

// ---------------- fused MLP: h = relu(x@W1+b1)@W2+b2 via f32 WMMA ----------------
// One workgroup = 4 waves = 16 rows of x. Stage1 result (16x256) parked in LDS.
__global__ __launch_bounds__(128) void mlp_wmma_kernel(
    const float* __restrict__ x, const float* __restrict__ W1, const float* __restrict__ b1,
    const float* __restrict__ W2, const float* __restrict__ b2, float* __restrict__ h) {
    __shared__ float t[16 * LDS_STRIDE];

    const int wave  = threadIdx.x >> 5;
    const int lane  = threadIdx.x & 31;
    const int rowL  = lane & 15;        // row of A frag / col of B,C frags
    const int khalf = lane >> 4;        // 0: K={0,1}, 1: K={2,3} within a k-step
    const long long m0 = (long long)blockIdx.x * 16;

    // ---- stage 1: t[:, wave*64 .. wave*64+63] = relu(x_tile @ W1 + b1) ----
    v8f acc[4] = {v8f{}, v8f{}, v8f{}, v8f{}};
    const float* xrow = x + (m0 + rowL) * IN_DIM;
    for (int k0 = 0; k0 < IN_DIM; k0 += 4) {
        const int ka = k0 + 2 * khalf;
        v2f a = *(const v2f*)(xrow + ka);
#pragma unroll
        for (int s = 0; s < 4; ++s) {
            const int n0 = wave * 64 + s * 16;
            v2f b;
            b.x = W1[(long long)ka * HID + n0 + rowL];
            b.y = W1[(long long)(ka + 1) * HID + n0 + rowL];
            acc[s] = __builtin_amdgcn_wmma_f32_16x16x4_f32(
                false, a, false, b, (short)0, acc[s], false, false);
        }
    }
#pragma unroll
    for (int s = 0; s < 4; ++s) {
        const int n = wave * 64 + s * 16 + rowL;
        const float bias = b1[n];
#pragma unroll
        for (int v = 0; v < 8; ++v) {
            const int M = v + 8 * khalf;
            const float val = acc[s][v] + bias;
            t[M * LDS_STRIDE + n] = val > 0.f ? val : 0.f;
        }
    }
    __syncthreads();

    // ---- stage 2: h[:, wave*16 .. wave*16+15] = t @ W2 + b2 ----
    v8f acc2 = v8f{};
    const int n0 = wave * 16;
    for (int k0 = 0; k0 < HID; k0 += 4) {
        const int ka = k0 + 2 * khalf;
        v2f a = *(const v2f*)(&t[rowL * LDS_STRIDE + ka]);
        v2f b;
        b.x = W2[(long long)ka * OUT_DIM + n0 + rowL];
        b.y = W2[(long long)(ka + 1) * OUT_DIM + n0 + rowL];
        acc2 = __builtin_amdgcn_wmma_f32_16x16x4_f32(
            false, a, false, b, (short)0, acc2, false, false);
    }
    const int n = n0 + rowL;
    const float bias2 = b2[n];
#pragma unroll
    for (int v = 0; v < 8; ++v) {
        const int M = v + 8 * khalf;
        h[(m0 + M) * OUT_DIM + n] = acc2[v] + bias2;
    }
}

// ---------------- propagation ----------------
// out = gamma0 * h0 ; also zero the ping-pong buffer that the first scatter targets
__global__ void init_acc_kernel(float* __restrict__ out, const float* __restrict__ h0,
                                float* __restrict__ hB, const float* __restrict__ gamma, long long n) {
    long long i = (long long)blockIdx.x * blockDim.x + threadIdx.x;
    if (i < n) {
        out[i] = gamma[0] * h0[i];
        hB[i]  = 0.f;
    }
}

// 64 threads per edge: coalesced gather of h_cur[col], L2-resident atomic scatter to h_next[row]
__global__ void scatter_kernel(const long long* __restrict__ row, const long long* __restrict__ col,
                               const float* __restrict__ norm, const float* __restrict__ h_cur,
                               float* __restrict__ h_next) {
    long long idx = (long long)blockIdx.x * blockDim.x + threadIdx.x;
    int e = (int)(idx >> 6);
    int j = (int)(idx & 63);
    if (e >= N_EDGES) return;
    int r = (int)row[e];
    int c = (int)col[e];
    float v = norm[e] * h_cur[(long long)c * OUT_DIM + j];
    atomicAdd(&h_next[(long long)r * OUT_DIM + j], v);
}

// out += gamma[k] * h_next ; zero old h_cur so it can serve as next scatter target
__global__ void accum_kernel(float* __restrict__ out, const float* __restrict__ h_next,
                             float* __restrict__ h_cur_zero, const float* __restrict__ gamma,
                             int k, long long n) {
    long long i = (long long)blockIdx.x * blockDim.x + threadIdx.x;
    if (i < n) {
        out[i] += gamma[k] * h_next[i];
        h_cur_zero[i] = 0.f;
    }
}

extern "C" void kernel_launch(void* const* d_in, const int* in_sizes, int n_in,
                              void* d_out, int out_size, void* d_ws, size_t ws_size,
                              hipStream_t stream) {
    const float*     x     = (const float*)d_in[0];
    const long long* eidx  = (const long long*)d_in[1];   // int64 [2, E]
    const float*     W1    = (const float*)d_in[2];
    const float*     b1    = (const float*)d_in[3];
    const float*     W2    = (const float*)d_in[4];
    const float*     b2    = (const float*)d_in[5];
    const float*     alpha = (const float*)d_in[6];
    float* out = (float*)d_out;

    const long long* row = eidx;
    const long long* col = eidx + N_EDGES;

    // workspace layout (256B-aligned slabs)
    char* ws = (char*)d_ws;
    float* gamma = (float*)(ws);                                     // 11 floats
    float* deg   = (float*)(ws + 256);                               // N floats (becomes deg_inv_sqrt)
    float* norm  = (float*)(ws + 256 + 400384);                      // E floats
    float* hA    = (float*)(ws + 256 + 400384 + 6400000);            // N*64
    float* hB    = (float*)(ws + 256 + 400384 + 6400000 + 25600000); // N*64

    const long long nh = (long long)N_NODES * OUT_DIM;
    const int TB = 256;

    hipLaunchKernelGGL(gamma_kernel, dim3(1), dim3(32), 0, stream, alpha, gamma);

    hipLaunchKernelGGL(zero_f32, dim3((N_NODES + TB - 1) / TB), dim3(TB), 0, stream, deg, (long long)N_NODES);
    hipLaunchKernelGGL(degree_kernel, dim3((N_EDGES + TB - 1) / TB), dim3(TB), 0, stream, row, deg, N_EDGES);
    hipLaunchKernelGGL(dinvsqrt_kernel, dim3((N_NODES + TB - 1) / TB), dim3(TB), 0, stream, deg, N_NODES);
    hipLaunchKernelGGL(norm_kernel, dim3((N_EDGES + TB - 1) / TB), dim3(TB), 0, stream, row, col, deg, norm, N_EDGES);

    // fused WMMA MLP -> hA
    hipLaunchKernelGGL(mlp_wmma_kernel, dim3(N_NODES / 16), dim3(128), 0, stream, x, W1, b1, W2, b2, hA);

    // acc0 = gamma[0]*h ; zero hB
    hipLaunchKernelGGL(init_acc_kernel, dim3((int)((nh + TB - 1) / TB)), dim3(TB), 0, stream,
                       out, hA, hB, gamma, nh);

    float* h_cur = hA;
    float* h_next = hB;
    const long long sc_threads = (long long)N_EDGES * OUT_DIM;
    for (int k = 1; k <= K_STEPS; ++k) {
        hipLaunchKernelGGL(scatter_kernel, dim3((int)((sc_threads + TB - 1) / TB)), dim3(TB), 0, stream,
                           row, col, norm, h_cur, h_next);
        hipLaunchKernelGGL(accum_kernel, dim3((int)((nh + TB - 1) / TB)), dim3(TB), 0, stream,
                           out, h_next, h_cur, gamma, k, nh);
        float* tmp = h_cur; h_cur = h_next; h_next = tmp;
    }
}